// NeighborhoodAttention3D_32203664785486
// MI455X (gfx1250) — compile-verified
//
#include <hip/hip_runtime.h>
#include <math.h>

// ---------------------------------------------------------------------------
// Problem constants (match the JAX reference)
// ---------------------------------------------------------------------------
#define BX   8
#define BY   28
#define BZ   28
#define NTOK (BX * BY * BZ)   // 6272 = 196 * 32
#define DIM  256
#define NH   8
#define DH   32
#define KX   3
#define KY   5
#define KZ   5
#define NNBR (KX * KY * KZ)   // 75

typedef float v2f __attribute__((ext_vector_type(2)));
typedef float v8f __attribute__((ext_vector_type(8)));

// ---------------------------------------------------------------------------
// WMMA fp32 GEMM with fused bias:  C[M,N] = A[M,K] @ B[K,N] + bias[N]
// N and K are compile-time so all B-fragment addresses strength-reduce to
// immediate offsets off one incrementing pointer (no per-iteration 64-bit
// multiplies in the hot loop).
//
// One wave computes a 32(M) x 64(N) tile via V_WMMA_F32_16X16X4_F32:
// 2 A fragments shared across 4 B fragments -> 8 WMMAs per 10 loads/k-step.
//
// 32-bit A 16x4 layout (ISA 7.12.2): lanes 0-15: M=lane, K={0,1} in v0,v1;
//                                    lanes 16-31: M=lane-16, K={2,3}.
// B 4x16 mirrors with N striped across lanes.
// C/D 16x16 layout: VGPR r, lanes 0-15 -> M=r, lanes 16-31 -> M=r+8.
// ---------------------------------------------------------------------------
template <int N, int K>
__global__ __launch_bounds__(128)
void na3d_wmma_gemm_bias(const float* __restrict__ A,
                         const float* __restrict__ B,
                         const float* __restrict__ bias,
                         float* __restrict__ C,
                         int M) {
  const int lane = threadIdx.x;        // 0..31
  const int half = lane >> 4;          // 0 or 1
  const int mr   = lane & 15;          // 0..15
  const int mBase = blockIdx.x * 32;
  const int nBase = blockIdx.y * 256 + threadIdx.y * 64;
  if (mBase >= M || nBase >= N) return;   // wave-uniform: EXEC stays all-1s

  v8f acc00 = {}, acc01 = {}, acc02 = {}, acc03 = {};
  v8f acc10 = {}, acc11 = {}, acc12 = {}, acc13 = {};

  const float* aRow0 = A + (size_t)(mBase + mr) * K + 2 * half;
  const float* aRow1 = aRow0 + (size_t)16 * K;
  // B row pointer for this lane: row (2*half), column nBase+mr; advances 4 rows/step
  const float* bp = B + (size_t)(2 * half) * N + nBase + mr;

  #pragma unroll 2
  for (int kb = 0; kb < K; kb += 4) {
    // A fragments: two consecutive K values for each of the two M-subtiles
    v2f a0 = *(const v2f*)(aRow0 + kb);
    v2f a1 = *(const v2f*)(aRow1 + kb);

    // B fragments: rows kb+2*half and kb+2*half+1, 4 N-subtiles of 16
    v2f b0, b1, b2, b3;
    b0.x = bp[0];      b0.y = bp[N];
    b1.x = bp[16];     b1.y = bp[N + 16];
    b2.x = bp[32];     b2.y = bp[N + 32];
    b3.x = bp[48];     b3.y = bp[N + 48];
    bp += 4 * N;

    acc00 = __builtin_amdgcn_wmma_f32_16x16x4_f32(false, a0, false, b0, (short)0, acc00, false, false);
    acc10 = __builtin_amdgcn_wmma_f32_16x16x4_f32(false, a1, false, b0, (short)0, acc10, false, false);
    acc01 = __builtin_amdgcn_wmma_f32_16x16x4_f32(false, a0, false, b1, (short)0, acc01, false, false);
    acc11 = __builtin_amdgcn_wmma_f32_16x16x4_f32(false, a1, false, b1, (short)0, acc11, false, false);
    acc02 = __builtin_amdgcn_wmma_f32_16x16x4_f32(false, a0, false, b2, (short)0, acc02, false, false);
    acc12 = __builtin_amdgcn_wmma_f32_16x16x4_f32(false, a1, false, b2, (short)0, acc12, false, false);
    acc03 = __builtin_amdgcn_wmma_f32_16x16x4_f32(false, a0, false, b3, (short)0, acc03, false, false);
    acc13 = __builtin_amdgcn_wmma_f32_16x16x4_f32(false, a1, false, b3, (short)0, acc13, false, false);
  }

  // Fused bias + store.  Element r of acc: row base + r + 8*half, col nBase+nt*16+mr.
  v8f accs[2][4] = {{acc00, acc01, acc02, acc03}, {acc10, acc11, acc12, acc13}};
  #pragma unroll
  for (int mt = 0; mt < 2; ++mt) {
    const int rowBase = mBase + mt * 16 + 8 * half;
    #pragma unroll
    for (int nt = 0; nt < 4; ++nt) {
      const int n = nBase + nt * 16 + mr;
      const float bi = bias[n];
      float* cp = C + (size_t)rowBase * N + n;
      #pragma unroll
      for (int r = 0; r < 8; ++r) {
        cp[(size_t)r * N] = accs[mt][nt][r] + bi;
      }
    }
  }
}

// ---------------------------------------------------------------------------
// Neighborhood attention core: one wave per (token, head), one block per token.
// Neighbor row offsets are computed once per block into LDS (all 8 head-waves
// share the token). Lane d owns channel d (DH == 32 == wave32): coalesced
// 128B K/V row loads, wave shuffle reductions for dots / softmax, softmax
// weights parked in LDS for uniform-address broadcast reads in the AV pass.
// qkv layout per token: [0..255]=Q, [256..511]=K, [512..767]=V, head h at h*32.
// ---------------------------------------------------------------------------
__global__ __launch_bounds__(256)
void na3d_attn(const float* __restrict__ qkv, float* __restrict__ aout) {
  __shared__ int   nbrOff[NNBR];      // neighbor token * 768 (qkv row offset)
  __shared__ float wLDS[NH][80];      // normalized softmax weights per head

  const int t    = blockIdx.x;      // token 0..6271
  const int h    = threadIdx.y;     // head 0..7
  const int lane = threadIdx.x;     // channel d 0..31

  if (h == 0) {
    const int z = t % BZ;
    const int y = (t / BZ) % BY;
    const int x = t / (BZ * BY);
    const int sx = min(max(x - KX / 2, 0), BX - KX);
    const int sy = min(max(y - KY / 2, 0), BY - KY);
    const int sz = min(max(z - KZ / 2, 0), BZ - KZ);
    for (int n = lane; n < NNBR; n += 32) {
      const int nx = n / (KY * KZ);
      const int rr = n % (KY * KZ);
      const int tn = ((sx + nx) * BY + (sy + rr / KZ)) * BZ + (sz + rr % KZ);
      nbrOff[n] = tn * 768;
    }
  }
  __syncthreads();

  const float scale = 0.17677669529663687f;   // 1/sqrt(32)
  const int   hoff  = h * DH;
  const float q = qkv[(size_t)t * 768 + hoff + lane] * scale;
  const float* kbase = qkv + 256 + hoff + lane;
  const float* vbase = qkv + 512 + hoff + lane;

  // ---- scores: 75 neighbors -> parked in 3 VGPRs across lanes -------------
  float sc0 = -INFINITY, sc1 = -INFINITY, sc2 = -INFINITY;
  #pragma unroll 5
  for (int n = 0; n < NNBR; ++n) {
    float p = q * kbase[nbrOff[n]];
    #pragma unroll
    for (int off = 16; off >= 1; off >>= 1) p += __shfl_xor(p, off, 32);
    if (lane == (n & 31)) {
      if (n < 32)      sc0 = p;
      else if (n < 64) sc1 = p;
      else             sc2 = p;
    }
  }

  // ---- softmax over 75 entries (sc2 valid only for lanes 0..10) -----------
  float m = fmaxf(sc0, fmaxf(sc1, sc2));
  #pragma unroll
  for (int off = 16; off >= 1; off >>= 1) m = fmaxf(m, __shfl_xor(m, off, 32));
  const float e0 = __expf(sc0 - m);
  const float e1 = __expf(sc1 - m);
  const float e2 = (lane < NNBR - 64) ? __expf(sc2 - m) : 0.0f;
  float s = e0 + e1 + e2;
  #pragma unroll
  for (int off = 16; off >= 1; off >>= 1) s += __shfl_xor(s, off, 32);
  const float inv = 1.0f / s;

  // Park normalized weights in LDS (same-wave LDS RAW is in-order: no barrier)
  wLDS[h][lane]      = e0 * inv;
  wLDS[h][lane + 32] = e1 * inv;
  if (lane < NNBR - 64) wLDS[h][lane + 64] = e2 * inv;

  // ---- output: sum_n attn_n * V_n[d] --------------------------------------
  float acc = 0.0f;
  #pragma unroll 5
  for (int n = 0; n < NNBR; ++n) {
    acc += wLDS[h][n] * vbase[nbrOff[n]];
  }
  aout[(size_t)t * DIM + hoff + lane] = acc;
}

// ---------------------------------------------------------------------------
// Launch: QKV GEMM -> attention -> projection GEMM
// ---------------------------------------------------------------------------
extern "C" void kernel_launch(void* const* d_in, const int* in_sizes, int n_in,
                              void* d_out, int out_size, void* d_ws, size_t ws_size,
                              hipStream_t stream) {
  const float* x      = (const float*)d_in[0];   // [6272, 256]
  const float* w_qkv  = (const float*)d_in[1];   // [256, 768]
  const float* b_qkv  = (const float*)d_in[2];   // [768]
  const float* w_proj = (const float*)d_in[3];   // [256, 256]
  const float* b_proj = (const float*)d_in[4];   // [256]
  float* out = (float*)d_out;                    // [6272, 256]

  float* qkv  = (float*)d_ws;                        // 6272*768 f32 = 19.27 MB
  float* aout = qkv + (size_t)NTOK * 768;            // 6272*256 f32 =  6.42 MB

  const dim3 blk(32, 4);   // 4 waves, each owns a 32x64 tile

  // 1) QKV projection: [6272,256] @ [256,768] + b   -> qkv
  na3d_wmma_gemm_bias<768, 256><<<dim3(NTOK / 32, 768 / 256), blk, 0, stream>>>(
      x, w_qkv, b_qkv, qkv, NTOK);

  // 2) Neighborhood attention: one wave per (token, head)
  na3d_attn<<<dim3(NTOK), dim3(32, NH), 0, stream>>>(qkv, aout);

  // 3) Output projection: [6272,256] @ [256,256] + b -> out
  na3d_wmma_gemm_bias<256, 256><<<dim3(NTOK / 32, 1), blk, 0, stream>>>(
      aout, w_proj, b_proj, out, NTOK);
}